// CodeGNN_10445360464521
// MI455X (gfx1250) — compile-verified
//
#include <hip/hip_runtime.h>

#define NNODES 50000
#define NEDGES 800000
#define FEATD  128
#define HID    256
#define EDIM   6
#define EHIDD  64
#define NGRAPH 64
#define LNEPS  1e-5f

typedef __attribute__((ext_vector_type(2))) float v2f;
typedef __attribute__((ext_vector_type(8))) float v8f;

__device__ __forceinline__ void atomicAddF(float* p, float v) {
  __hip_atomic_fetch_add(p, v, __ATOMIC_RELAXED, __HIP_MEMORY_SCOPE_AGENT);
}

__device__ __forceinline__ v8f wmma_f32(v2f a, v2f b, v8f c) {
  return __builtin_amdgcn_wmma_f32_16x16x4_f32(
      /*neg_a=*/false, a, /*neg_b=*/false, b,
      /*c_mod=*/(short)0, c, /*reuse_a=*/false, /*reuse_b=*/false);
}

// ---------------------------------------------------------------------------
// WMMA fp32 GEMM, compile-time K/N so W/A addressing folds into immediate
// offsets:  C[M,N] = A[M,K] @ W[K,N] (+bias[col]) (+addC) (+rowscale*bias2)
// Each wave computes a 32x32 macro-tile = 2x2 of 16x16 V_WMMA_F32_16X16X4_F32
// tiles (4 WMMA per k-step sharing 2 A-frags + 2 B-frags).
// M ragged: A row indices clamped (EXEC must stay all-ones for WMMA),
// stores row-guarded.
// Lane layouts per CDNA5 ISA 7.12.2:
//   A (16x4):  lanes 0-15 rows 0-15; VGPR0={K0|K2}, VGPR1={K1|K3}
//   B (4x16):  row K striped across lanes within a VGPR (same half split)
//   C/D:       VGPR r -> row r (lanes 0-15), row r+8 (lanes 16-31)
// ---------------------------------------------------------------------------
template <int K, int N>
__global__ void wmma_gemm_kernel(const float* __restrict__ A,
                                 const float* __restrict__ W,
                                 const float* __restrict__ bias,
                                 const float* __restrict__ addC,
                                 const float* __restrict__ rowscale,
                                 const float* __restrict__ bias2,
                                 float* __restrict__ C,
                                 int M) {
  const int lane = threadIdx.x & 31;
  const int wid  = (blockIdx.x * blockDim.x + threadIdx.x) >> 5;
  constexpr int NT2 = N / 32;             // macro n-tiles
  const int mtiles = (M + 31) >> 5;       // macro m-tiles (32 rows each)
  const int mt = wid / NT2;
  const int nt = wid - mt * NT2;
  if (mt >= mtiles) return;               // wave-uniform

  const int halfsel = lane >> 4;          // 0 or 1
  const int l16     = lane & 15;
  const int kofs    = halfsel * 2;

  const int row0 = mt * 32 + l16;         // m-subtile 0 A row
  const int row1 = row0 + 16;             // m-subtile 1 A row
  const int r0c  = min(row0, M - 1);      // clamp: keep EXEC uniform
  const int r1c  = min(row1, M - 1);
  const int col0 = nt * 32 + l16;         // n-subtile 0 B col
  // n-subtile 1 col = col0 + 16

  const float* Ap0 = A + (size_t)r0c * K + kofs;
  const float* Ap1 = A + (size_t)r1c * K + kofs;
  const float* Wp  = W + (size_t)kofs * N + col0;  // B(k+kofs, col0) at Wp[k*N]

  v8f acc00 = {}, acc01 = {}, acc10 = {}, acc11 = {};
  #pragma unroll 4
  for (int k = 0; k < K; k += 4) {
    v2f a0; a0.x = Ap0[k];          a0.y = Ap0[k + 1];
    v2f a1; a1.x = Ap1[k];          a1.y = Ap1[k + 1];
    v2f b0; b0.x = Wp[k * N];       b0.y = Wp[k * N + N];
    v2f b1; b1.x = Wp[k * N + 16];  b1.y = Wp[k * N + N + 16];
    acc00 = wmma_f32(a0, b0, acc00);
    acc01 = wmma_f32(a0, b1, acc01);
    acc10 = wmma_f32(a1, b0, acc10);
    acc11 = wmma_f32(a1, b1, acc11);
  }

  // ---- epilogue ----
  const int rbase0 = mt * 32 + halfsel * 8;        // rows for acc0x
  const int rbase1 = rbase0 + 16;                  // rows for acc1x
  #pragma unroll
  for (int r = 0; r < 8; ++r) {
    #pragma unroll
    for (int t = 0; t < 4; ++t) {                  // t = {m-sub, n-sub}
      const int row = ((t & 2) ? rbase1 : rbase0) + r;
      const int col = col0 + ((t & 1) ? 16 : 0);
      if (row < M) {
        const size_t idx = (size_t)row * N + col;
        float v = (t == 0) ? acc00[r] : (t == 1) ? acc01[r] : (t == 2) ? acc10[r] : acc11[r];
        if (bias)     v += bias[col];
        if (addC)     v += addC[idx];
        if (rowscale) v += rowscale[row] * bias2[col];
        C[idx] = v;
      }
    }
  }
}

// ---------------------------------------------------------------------------
// Edge encoder + dst-segment reduction: segEf[dst] += ea[e]@edge_w + edge_b,
// deg[dst] += 1.  64 threads per edge (one per EHID output).
// ---------------------------------------------------------------------------
__global__ void edge_enc_kernel(const float* __restrict__ ea,
                                const int* __restrict__ ei,
                                const float* __restrict__ ew,
                                const float* __restrict__ eb,
                                float* __restrict__ segEf,
                                float* __restrict__ degv) {
  const int gid = blockIdx.x * blockDim.x + threadIdx.x;
  const int e = gid >> 6;
  const int j = gid & 63;
  if (e >= NEDGES) return;
  const int dst = ei[NEDGES + e];
  float acc = eb[j];
  #pragma unroll
  for (int k = 0; k < EDIM; ++k) acc += ea[(size_t)e * EDIM + k] * ew[k * EHIDD + j];
  atomicAddF(&segEf[(size_t)dst * EHIDD + j], acc);
  if (j == 0) atomicAddF(&degv[dst], 1.0f);
}

// Message scatter: agg[dst] += hl[src], 256 floats per edge.
__global__ void scatter_kernel(const float* __restrict__ hl,
                               const int* __restrict__ ei,
                               float* __restrict__ agg) {
  const int gid = blockIdx.x * blockDim.x + threadIdx.x;
  const int e = gid >> 6;
  if (e >= NEDGES) return;
  const int c = (gid & 63) << 2;
  const int s = ei[e];
  const int d = ei[NEDGES + e];
  const float4 v = *(const float4*)(hl + (size_t)s * HID + c);
  float* p = agg + (size_t)d * HID + c;
  atomicAddF(p + 0, v.x);
  atomicAddF(p + 1, v.y);
  atomicAddF(p + 2, v.z);
  atomicAddF(p + 3, v.w);
}

// out = relu(agg) + hself   (elementwise)
__global__ void combine_kernel(const float* __restrict__ agg,
                               const float* __restrict__ hself,
                               float* __restrict__ out, int n) {
  const int i = blockIdx.x * blockDim.x + threadIdx.x;
  if (i >= n) return;
  out[i] = fmaxf(agg[i], 0.0f) + hself[i];
}

// Row-wise LayerNorm (in place) + optional ReLU. One wave32 per row.
__global__ void ln_kernel(float* __restrict__ x,
                          const float* __restrict__ g,
                          const float* __restrict__ b,
                          int D, int relu) {
  const int row = blockIdx.x;
  const int lane = threadIdx.x;
  float* rp = x + (size_t)row * D;
  float s = 0.f, sq = 0.f;
  for (int j = lane; j < D; j += 32) { const float v = rp[j]; s += v; sq += v * v; }
  #pragma unroll
  for (int o = 16; o > 0; o >>= 1) { s += __shfl_xor(s, o, 32); sq += __shfl_xor(sq, o, 32); }
  const float m = s / (float)D;
  const float var = sq / (float)D - m * m;
  const float r = rsqrtf(var + LNEPS);
  for (int j = lane; j < D; j += 32) {
    float v = (rp[j] - m) * r * g[j] + b[j];
    if (relu) v = fmaxf(v, 0.0f);
    rp[j] = v;
  }
}

// Global mean-pool accumulation: pooled[batch[n]] += h[n]; cnt[batch[n]] += 1
__global__ void pool_kernel(const float* __restrict__ h,
                            const int* __restrict__ batch,
                            float* __restrict__ pooled,
                            float* __restrict__ cnt) {
  const int gid = blockIdx.x * blockDim.x + threadIdx.x;
  const int n = gid >> 8;
  const int d = gid & 255;
  if (n >= NNODES) return;
  const int gidx = batch[n];
  atomicAddF(&pooled[(size_t)gidx * HID + d], h[(size_t)n * HID + d]);
  if (d == 0) atomicAddF(&cnt[gidx], 1.0f);
}

__global__ void mean_kernel(const float* __restrict__ pooled,
                            const float* __restrict__ cnt,
                            float* __restrict__ g) {
  const int row = blockIdx.x;
  const int col = threadIdx.x;
  g[(size_t)row * HID + col] = pooled[(size_t)row * HID + col] / fmaxf(cnt[row], 1.0f);
}

// Final 2-class projection: C[row,col] = A[row,:] @ W[:,col] + b (tiny)
__global__ void fc_kernel(const float* __restrict__ A, const float* __restrict__ W,
                          const float* __restrict__ b, float* __restrict__ C,
                          int K, int N) {
  const int row = blockIdx.x;
  const int col = threadIdx.x;
  if (col >= N) return;
  float acc = b[col];
  for (int k = 0; k < K; ++k) acc += A[(size_t)row * K + k] * W[(size_t)k * N + col];
  C[(size_t)row * N + col] = acc;
}

// ---------------------------------------------------------------------------
extern "C" void kernel_launch(void* const* d_in, const int* in_sizes, int n_in,
                              void* d_out, int out_size, void* d_ws, size_t ws_size,
                              hipStream_t stream) {
  (void)in_sizes; (void)n_in; (void)out_size; (void)ws_size;

  const float* x     = (const float*)d_in[0];
  const int*   ei    = (const int*)d_in[1];   // [2, E]: src row, then dst row
  const float* ea    = (const float*)d_in[2];
  const int*   batch = (const int*)d_in[3];
  #define P(i) ((const float*)d_in[(i)])
  const float* enc_w = P(4),  *enc_b = P(5),  *enc_g = P(6),  *enc_bt = P(7);
  const float* edge_w = P(8), *edge_b = P(9);
  // layer i base = 10 + 8*i: W, b, Ws, bs, We, be, g, bt
  // sk: 42..47 ; c1: 48..51 ; c2: 52..55 ; c3: 56..57

  // ---- workspace carve ----
  char* ws = (char*)d_ws;
  auto carve = [&](size_t bytes) -> float* {
    float* p = (float*)ws;
    ws += (bytes + 255) & ~(size_t)255;
    return p;
  };
  const size_t NHB = (size_t)NNODES * HID * sizeof(float);
  float* h      = carve(NHB);
  float* hl     = carve(NHB);
  float* hself  = carve(NHB);
  float* agg    = carve(NHB);
  float* segEf  = carve((size_t)NNODES * EHIDD * sizeof(float));
  float* degv   = carve((size_t)NNODES * sizeof(float));
  float* pooled = carve((size_t)NGRAPH * HID * sizeof(float));
  float* cnt    = carve((size_t)NGRAPH * sizeof(float));
  float* gA     = carve((size_t)NGRAPH * HID * sizeof(float));
  float* gB     = carve((size_t)NGRAPH * HID * sizeof(float));
  float* gC     = carve((size_t)NGRAPH * (HID / 2) * sizeof(float));

  auto gemm = [&](const float* A, const float* Wt, const float* bias,
                  const float* addC, const float* rs, const float* b2,
                  float* Cp, int M, int K, int N) {
    const int waves = ((M + 31) / 32) * (N / 32);
    const int wpb = 8;                              // 256-thread blocks
    const int grid = (waves + wpb - 1) / wpb;
    if (K == 128 && N == 256)
      wmma_gemm_kernel<128, 256><<<grid, 256, 0, stream>>>(A, Wt, bias, addC, rs, b2, Cp, M);
    else if (K == 256 && N == 256)
      wmma_gemm_kernel<256, 256><<<grid, 256, 0, stream>>>(A, Wt, bias, addC, rs, b2, Cp, M);
    else if (K == 64 && N == 256)
      wmma_gemm_kernel<64, 256><<<grid, 256, 0, stream>>>(A, Wt, bias, addC, rs, b2, Cp, M);
    else if (K == 256 && N == 128)
      wmma_gemm_kernel<256, 128><<<grid, 256, 0, stream>>>(A, Wt, bias, addC, rs, b2, Cp, M);
  };

  // ---- zero atomic accumulators (replay-safe) ----
  hipMemsetAsync(segEf, 0, (size_t)NNODES * EHIDD * sizeof(float), stream);
  hipMemsetAsync(degv,  0, (size_t)NNODES * sizeof(float), stream);
  hipMemsetAsync(pooled,0, (size_t)NGRAPH * HID * sizeof(float), stream);
  hipMemsetAsync(cnt,   0, (size_t)NGRAPH * sizeof(float), stream);

  // ---- node encoder: h = relu(LN(x @ enc_w + enc_b)) ----
  gemm(x, enc_w, enc_b, nullptr, nullptr, nullptr, h, NNODES, FEATD, HID);
  ln_kernel<<<NNODES, 32, 0, stream>>>(h, enc_g, enc_bt, HID, 1);

  // ---- edge encoder + dst-segment reduction (layer-invariant) ----
  {
    const int tot = NEDGES * 64;
    edge_enc_kernel<<<(tot + 255) / 256, 256, 0, stream>>>(ea, ei, edge_w, edge_b, segEf, degv);
  }

  // ---- GNN layers ----
  for (int i = 0; i < 4; ++i) {
    const int base = 10 + 8 * i;
    const float *Wl = P(base + 0), *bl = P(base + 1);
    const float *Ws = P(base + 2), *bs = P(base + 3);
    const float *We = P(base + 4), *be = P(base + 5);
    const float *lg = P(base + 6), *lbt = P(base + 7);

    gemm(h, Wl, bl, nullptr, nullptr, nullptr, hl, NNODES, HID, HID);
    gemm(h, Ws, bs, nullptr, nullptr, nullptr, hself, NNODES, HID, HID);
    if (i > 0) {  // fold skip projection in: hself += h @ sk_w + sk_b
      const float* skw = P(42 + 2 * (i - 1));
      const float* skb = P(43 + 2 * (i - 1));
      gemm(h, skw, skb, hself, nullptr, nullptr, hself, NNODES, HID, HID);
    }
    // agg = hl (self loop) + segEf @ We + deg * be
    gemm(segEf, We, nullptr, hl, degv, be, agg, NNODES, EHIDD, HID);
    // agg += segment_sum(hl[src], dst)
    {
      const int tot = NEDGES * 64;
      scatter_kernel<<<(tot + 255) / 256, 256, 0, stream>>>(hl, ei, agg);
    }
    // h = LN(relu(agg) + hself) [+ relu except last layer]
    {
      const int n = NNODES * HID;
      combine_kernel<<<(n + 255) / 256, 256, 0, stream>>>(agg, hself, h, n);
    }
    ln_kernel<<<NNODES, 32, 0, stream>>>(h, lg, lbt, HID, (i < 3) ? 1 : 0);
  }

  // ---- global mean pool ----
  {
    const int tot = NNODES * HID;
    pool_kernel<<<(tot + 255) / 256, 256, 0, stream>>>(h, batch, pooled, cnt);
  }
  mean_kernel<<<NGRAPH, HID, 0, stream>>>(pooled, cnt, gA);

  // ---- classifier head (c1/c2 on WMMA, c3 scalar) ----
  gemm(gA, P(48), P(49), nullptr, nullptr, nullptr, gB, NGRAPH, HID, HID);
  ln_kernel<<<NGRAPH, 32, 0, stream>>>(gB, P(50), P(51), HID, 1);
  gemm(gB, P(52), P(53), nullptr, nullptr, nullptr, gC, NGRAPH, HID, HID / 2);
  ln_kernel<<<NGRAPH, 32, 0, stream>>>(gC, P(54), P(55), HID / 2, 1);
  fc_kernel<<<NGRAPH, 2, 0, stream>>>(gC, P(56), P(57), (float*)d_out, HID / 2, 2);
  #undef P
}